// GCNLinkPredictor_65000035058075
// MI455X (gfx1250) — compile-verified
//
#include <hip/hip_runtime.h>
#include <hip/hip_bf16.h>

typedef __attribute__((ext_vector_type(2))) float v2f;
typedef __attribute__((ext_vector_type(4))) float v4f;
typedef __attribute__((ext_vector_type(8))) float v8f;

#define NNODES    50000
#define NEDGES    640000
#define CH        128
#define ROW_TILES 3125   // NNODES / 16 exactly

// ---------------------------------------------------------------- degree prep
__global__ void gcn_deg_init(float* __restrict__ deg) {
    int i = blockIdx.x * blockDim.x + threadIdx.x;
    if (i < NNODES) deg[i] = 1.0f;           // self-loop contribution
}

__global__ void gcn_deg_accum(const int* __restrict__ dst, float* __restrict__ deg) {
    int e = blockIdx.x * blockDim.x + threadIdx.x;
    if (e < NEDGES) unsafeAtomicAdd(&deg[dst[e]], 1.0f);
}

__global__ void gcn_dinv(const float* __restrict__ deg, float* __restrict__ dinv) {
    int i = blockIdx.x * blockDim.x + threadIdx.x;
    if (i < NNODES) {
        float d = deg[i];
        dinv[i] = (d > 0.0f) ? rsqrtf(d) : 0.0f;
    }
}

// ---------------------------------------------------------------- WMMA GEMM
// H[N,128] = X[N,128] @ W[128,128], fp32 via V_WMMA_F32_16X16X4_F32.
// 8 waves / block; wave w computes rows [rt*16, rt*16+16) x all 128 cols.
__global__ __launch_bounds__(256) void gcn_gemm_wmma(const float* __restrict__ X,
                                                     const float* __restrict__ W,
                                                     float* __restrict__ H) {
    __shared__ float Wt[CH * CH];   // 64 KB: Wt[c*128 + k] = W[k*128 + c]
    for (int idx = threadIdx.x; idx < CH * CH; idx += 256) {
        int k = idx >> 7, c = idx & 127;
        Wt[c * CH + k] = W[idx];
    }
    __syncthreads();

    const int wave = threadIdx.x >> 5;
    const int lane = threadIdx.x & 31;
    const int rt   = blockIdx.x * 8 + wave;
    if (rt >= ROW_TILES) return;              // wave-uniform: EXEC stays all-1s

    const int half = lane >> 4;               // 0: K={0,1}, 1: K={2,3} of each chunk
    const int mn   = lane & 15;               // A: row within tile / B: col within tile

    const float* __restrict__ xrow = X + (size_t)(rt * 16 + mn) * CH;

    v8f acc[8];
    v8f zero = {};
    #pragma unroll
    for (int c = 0; c < 8; ++c) acc[c] = zero;

    #pragma unroll 4
    for (int kk = 0; kk < CH; kk += 4) {
        // A fragment: X[rt*16+mn][kk + 2*half .. +1]
        v2f a = *(const v2f*)(xrow + kk + 2 * half);
        #pragma unroll
        for (int c = 0; c < 8; ++c) {
            // B fragment: W[kk + 2*half .. +1][c*16+mn] == Wt[(c*16+mn)*128 + kk+2*half ..]
            v2f b = *(const v2f*)(&Wt[(c * 16 + mn) * CH + kk + 2 * half]);
            acc[c] = __builtin_amdgcn_wmma_f32_16x16x4_f32(
                false, a, false, b, (short)0, acc[c], false, false);
        }
    }

    // C layout: VGPR i -> row (half*8 + i), col mn within each 16-col tile
    #pragma unroll
    for (int c = 0; c < 8; ++c) {
        #pragma unroll
        for (int i = 0; i < 8; ++i) {
            H[(size_t)(rt * 16 + half * 8 + i) * CH + c * 16 + mn] = acc[c][i];
        }
    }
}

// ------------------------------------------------- aggregation (self loop + bias)
__global__ void gcn_agg_init(const float* __restrict__ H, const float* __restrict__ dinv,
                             const float* __restrict__ bias, float* __restrict__ out) {
    int idx = blockIdx.x * blockDim.x + threadIdx.x;     // over NNODES*32
    if (idx >= NNODES * 32) return;
    int node = idx >> 5;
    int ch   = (idx & 31) * 4;
    float s  = dinv[node];
    s = s * s;                                           // self-loop norm
    v4f h = *(const v4f*)(&H[(size_t)node * CH + ch]);
    v4f b = *(const v4f*)(&bias[ch]);
    v4f o = h * s + b;
    *(v4f*)(&out[(size_t)node * CH + ch]) = o;
}

// one wave per edge; each lane handles 4 channels
__global__ __launch_bounds__(256) void gcn_agg_edges(const int* __restrict__ src,
                                                     const int* __restrict__ dst,
                                                     const float* __restrict__ dinv,
                                                     const float* __restrict__ H,
                                                     float* __restrict__ out) {
    int e = blockIdx.x * 8 + (threadIdx.x >> 5);
    if (e >= NEDGES) return;
    int lane = threadIdx.x & 31;
    int s = src[e];
    int d = dst[e];
    float norm = dinv[s] * dinv[d];
    int ch = lane * 4;
    v4f h = *(const v4f*)(&H[(size_t)s * CH + ch]);
    float* op = &out[(size_t)d * CH + ch];
    unsafeAtomicAdd(op + 0, h.x * norm);
    unsafeAtomicAdd(op + 1, h.y * norm);
    unsafeAtomicAdd(op + 2, h.z * norm);
    unsafeAtomicAdd(op + 3, h.w * norm);
}

__global__ void gcn_relu(float* __restrict__ x, int n4) {
    int i = blockIdx.x * blockDim.x + threadIdx.x;
    if (i >= n4) return;
    v4f v = *(const v4f*)(&x[i * 4]);
    v.x = fmaxf(v.x, 0.0f); v.y = fmaxf(v.y, 0.0f);
    v.z = fmaxf(v.z, 0.0f); v.w = fmaxf(v.w, 0.0f);
    *(v4f*)(&x[i * 4]) = v;
}

// ---------------------------------------------------------------- launcher
extern "C" void kernel_launch(void* const* d_in, const int* in_sizes, int n_in,
                              void* d_out, int out_size, void* d_ws, size_t ws_size,
                              hipStream_t stream) {
    const float* x   = (const float*)d_in[0];
    const int*   ei  = (const int*)d_in[1];       // [2, NEDGES]
    const float* W1  = (const float*)d_in[2];
    const float* b1  = (const float*)d_in[3];
    const float* W2  = (const float*)d_in[4];
    const float* b2  = (const float*)d_in[5];
    float*       out = (float*)d_out;

    const int* src = ei;
    const int* dst = ei + NEDGES;

    float* ws   = (float*)d_ws;
    float* deg  = ws;                       // 50000
    float* dinv = ws + 65536;               // 50000
    float* H    = ws + 131072;              // 6,400,000
    float* out1 = ws + 131072 + 6400000;    // 6,400,000

    const int T = 256;
    // degrees
    gcn_deg_init <<<(NNODES + T - 1) / T, T, 0, stream>>>(deg);
    gcn_deg_accum<<<(NEDGES + T - 1) / T, T, 0, stream>>>(dst, deg);
    gcn_dinv     <<<(NNODES + T - 1) / T, T, 0, stream>>>(deg, dinv);

    const int gemm_blocks = (ROW_TILES + 7) / 8;          // 391
    const int agg_init_blocks = (NNODES * 32 + T - 1) / T; // 6250
    const int agg_edge_blocks = (NEDGES + 7) / 8;          // 80000
    const int relu_blocks = (NNODES * CH / 4 + T - 1) / T; // 6250

    // ---- layer 1
    gcn_gemm_wmma<<<gemm_blocks, T, 0, stream>>>(x, W1, H);
    gcn_agg_init <<<agg_init_blocks, T, 0, stream>>>(H, dinv, b1, out1);
    gcn_agg_edges<<<agg_edge_blocks, T, 0, stream>>>(src, dst, dinv, H, out1);
    gcn_relu     <<<relu_blocks, T, 0, stream>>>(out1, NNODES * CH / 4);

    // ---- layer 2
    gcn_gemm_wmma<<<gemm_blocks, T, 0, stream>>>(out1, W2, H);
    gcn_agg_init <<<agg_init_blocks, T, 0, stream>>>(H, dinv, b2, out);
    gcn_agg_edges<<<agg_edge_blocks, T, 0, stream>>>(src, dst, dinv, H, out);
}